// GraphConvolution_6854767804919
// MI455X (gfx1250) — compile-verified
//
#include <hip/hip_runtime.h>
#include <hip/hip_bf16.h>

// GraphConvolution: out = (A + A^T) @ (input @ weight) + bias, COO edges with
// per-edge relation scalar. fp32 throughout (matches reference dtype).
//
// Phase 1: out[i,:] = bias                         (init_bias_kernel)
// Phase 2: support = input @ weight  via WMMA f32  (gemm_wmma_kernel, d_ws)
// Phase 3: per-edge scatter with f32 atomics       (scatter_kernel)

typedef float v2f __attribute__((ext_vector_type(2)));
typedef float v4f __attribute__((ext_vector_type(4)));
typedef float v8f __attribute__((ext_vector_type(8)));

#define DIM 128

// ---------------------------------------------------------------- Phase 1
__global__ void gc_init_bias_kernel(float* __restrict__ out,
                                    const float* __restrict__ bias,
                                    int total) {
    int idx = blockIdx.x * blockDim.x + threadIdx.x;
    if (idx < total) out[idx] = bias[idx & (DIM - 1)];
}

// ---------------------------------------------------------------- Phase 2
// support[N,128] = input[N,128] @ weight[128,128]
// One wave computes one 16x16 output tile with V_WMMA_F32_16X16X4_F32.
// K=128 is processed in two chunks of 64: each chunk preloads 16 A fragments
// (global_load_b64, unconditional — tail rows are CLAMPED, not predicated,
// since rows m>=N only feed output rows m>=N which are never stored) and
// 16 B fragments, then issues 16 back-to-back WMMAs.
//
// ISA VGPR layouts (CDNA5 ISA):
//   A 16x4 (MxK):  lanes 0-15 -> M=lane,    vgpr0/1 = K=0,1
//                  lanes 16-31 -> M=lane-16, vgpr0/1 = K=2,3
//   B 4x16 (KxN):  mirrored: lanes 0-15 -> N=lane, vgpr0/1 = K=0,1
//                  lanes 16-31 -> N=lane-16, vgpr0/1 = K=2,3
//   C/D 16x16 f32: vgpr r -> M=r (lanes 0-15) / M=r+8 (lanes 16-31), N=lane%16
__global__ void gc_gemm_wmma_kernel(const float* __restrict__ A,  // [N,128]
                                    const float* __restrict__ W,  // [128,128]
                                    float* __restrict__ S,        // [N,128]
                                    int N) {
    const int wave_id = (blockIdx.x * blockDim.x + threadIdx.x) >> 5;  // uniform
    const int lane    = threadIdx.x & 31;
    const int tiles_n = DIM / 16;                  // 8 column tiles
    const int tm = wave_id / tiles_n;
    const int tn = wave_id % tiles_n;
    const int m0 = tm * 16;
    if (m0 >= N) return;                           // wave-uniform branch
    const int n0 = tn * 16;

    const int half = lane >> 4;                    // selects K pair within the 4
    const int lr   = lane & 15;

    int arow = m0 + lr;
    if (arow > N - 1) arow = N - 1;                // clamp (no predication needed)
    const float* __restrict__ arow_p = A + (size_t)arow * DIM + half * 2;
    const float* __restrict__ wcol_p = W + (size_t)(half * 2) * DIM + n0 + lr;

    v8f c = {0.f, 0.f, 0.f, 0.f, 0.f, 0.f, 0.f, 0.f};

#pragma unroll
    for (int chunk = 0; chunk < 2; ++chunk) {      // K = [0,64) then [64,128)
        const int kbase = chunk * 64;

        v2f a[16];
#pragma unroll
        for (int i = 0; i < 16; ++i)               // aligned 8B loads -> b64
            a[i] = *(const v2f*)(arow_p + kbase + i * 4);

        v2f b[16];
#pragma unroll
        for (int i = 0; i < 16; ++i) {
            const float* wp = wcol_p + (size_t)(kbase + i * 4) * DIM;
            b[i].x = wp[0];
            b[i].y = wp[DIM];
        }

#pragma unroll
        for (int i = 0; i < 16; ++i) {
            // (neg_a, A, neg_b, B, c_mod, C, reuse_a, reuse_b)
            c = __builtin_amdgcn_wmma_f32_16x16x4_f32(false, a[i], false, b[i],
                                                      (short)0, c, false, false);
        }
    }

#pragma unroll
    for (int r = 0; r < 8; ++r) {
        const int mrow = m0 + half * 8 + r;
        if (mrow < N) S[(size_t)mrow * DIM + n0 + lr] = c[r];
    }
}

// ---------------------------------------------------------------- Phase 3
// One wave per edge. Lane j owns floats [4j, 4j+4) of the 128-wide row.
// out[row] += alp * support[col];  out[col] += alp * support[row]
// support (51 MB) + out (51 MB) are L2-resident (192 MB L2), so the ~410M
// f32 atomics run at L2 atomic throughput, not HBM bandwidth.
__global__ void gc_scatter_kernel(const float* __restrict__ S,
                                  const int* __restrict__ edge_index,  // [2,E]
                                  const int* __restrict__ rel,         // [E]
                                  const float* __restrict__ alpha,     // [238]
                                  float* __restrict__ out,
                                  int E) {
    const int e    = (blockIdx.x * blockDim.x + threadIdx.x) >> 5;  // uniform
    const int lane = threadIdx.x & 31;
    if (e >= E) return;

    const int r  = edge_index[e];
    const int cl = edge_index[E + e];
    const int rl = rel[e];
    const float alp = (rl == 0) ? 0.f : alpha[rl];   // padding_idx=0 -> zero row
    if (alp == 0.f) return;                          // wave-uniform; adding 0 == skip

    const v4f sc = ((const v4f*)(S + (size_t)cl * DIM))[lane];
    const v4f sr = ((const v4f*)(S + (size_t)r  * DIM))[lane];
    float* __restrict__ orow = out + (size_t)r  * DIM + lane * 4;
    float* __restrict__ ocol = out + (size_t)cl * DIM + lane * 4;

#pragma unroll
    for (int t = 0; t < 4; ++t) {
        atomicAdd(orow + t, alp * sc[t]);
        atomicAdd(ocol + t, alp * sr[t]);
    }
}

// ---------------------------------------------------------------- launch
extern "C" void kernel_launch(void* const* d_in, const int* in_sizes, int n_in,
                              void* d_out, int out_size, void* d_ws, size_t ws_size,
                              hipStream_t stream) {
    const float* input      = (const float*)d_in[0];   // [N,128]
    const int*   edge_index = (const int*)  d_in[1];   // [2,E]
    const int*   rel        = (const int*)  d_in[2];   // [E]
    /* d_in[3] = n (device scalar; N derived from in_sizes instead) */
    const float* weight     = (const float*)d_in[4];   // [128,128]
    const float* alpha_emb  = (const float*)d_in[5];   // [238,1]
    const float* bias       = (const float*)d_in[6];   // [128]
    float* out     = (float*)d_out;
    float* support = (float*)d_ws;                     // [N,128] scratch

    const int N = in_sizes[0] / DIM;
    const int E = in_sizes[1] / 2;

    // Phase 1: out = bias (broadcast per row)
    const int total = N * DIM;
    gc_init_bias_kernel<<<(total + 255) / 256, 256, 0, stream>>>(out, bias, total);

    // Phase 2: support = input @ weight  (one wave per 16x16 tile, 8 waves/block)
    const int tiles = ((N + 15) / 16) * (DIM / 16);
    gc_gemm_wmma_kernel<<<(tiles + 7) / 8, 256, 0, stream>>>(input, weight, support, N);

    // Phase 3: edge scatter (one wave per edge, 8 edges/block)
    gc_scatter_kernel<<<(E + 7) / 8, 256, 0, stream>>>(support, edge_index, rel,
                                                       alpha_emb, out, E);
}